// AdditiveAttention_22600117911844
// MI455X (gfx1250) — compile-verified
//
#include <hip/hip_runtime.h>

#define B_   4
#define SQ_  512
#define SK_  512
#define H_   256
#define AD_  128

typedef __attribute__((ext_vector_type(2))) float v2f;
typedef __attribute__((ext_vector_type(8))) float v8f;

// ---------------------------------------------------------------------------
// One 16x16 fp32 output tile per wave, K-loop of V_WMMA_F32_16X16X4_F32.
// A layout (16x4):  lanes 0-15 = rows, VGPR0/1 = K {0,1}; lanes 16-31 = K {2,3}
// B layout (4x16):  lanes 0-15 = cols, VGPR0/1 = K {0,1}; lanes 16-31 = K {2,3}
// C/D layout:       reg r: lanes 0-15 -> row r, lanes 16-31 -> row r+8
// ---------------------------------------------------------------------------
__device__ __forceinline__ v8f wmma_f32_tile(const float* __restrict__ Abase, int lda,
                                             const float* __restrict__ Bbase, int ldb,
                                             int K, v8f c)
{
    const int lane = threadIdx.x & 31;
    const int idx  = lane & 15;          // row of A / col of B
    const int klo  = (lane >> 4) << 1;   // 0 or 2
    for (int k = 0; k < K; k += 4) {
        v2f a, b;
        a.x = Abase[idx * lda + (k + klo)];
        a.y = Abase[idx * lda + (k + klo + 1)];
        b.x = Bbase[(k + klo)     * ldb + idx];
        b.y = Bbase[(k + klo + 1) * ldb + idx];
        c = __builtin_amdgcn_wmma_f32_16x16x4_f32(false, a, false, b,
                                                  (short)0, c, false, false);
    }
    return c;
}

__device__ __forceinline__ void store_tile(float* __restrict__ Cbase, int ldc, v8f c)
{
    const int lane  = threadIdx.x & 31;
    const int col   = lane & 15;
    const int rbase = (lane >> 4) << 3;  // 0 or 8
#pragma unroll
    for (int r = 0; r < 8; ++r)
        Cbase[(rbase + r) * ldc + col] = c[r];
}

// ---------------------------------------------------------------------------
// Kernel 1: Qt = Q @ Wq, Kt = K @ Wk   (M=2048, N=128, K=256), fused via flag
// ---------------------------------------------------------------------------
__global__ __launch_bounds__(128) void k_project(
    const float* __restrict__ Q, const float* __restrict__ Kin,
    const float* __restrict__ Wq, const float* __restrict__ Wk,
    float* __restrict__ Qt, float* __restrict__ Kt)
{
    const int wave  = (blockIdx.x * blockDim.x + threadIdx.x) >> 5;
    const int which = wave >> 10;          // 0: Q-side, 1: K-side (1024 tiles each)
    const int t     = wave & 1023;
    const int mt    = t >> 3;              // 128 M-tiles
    const int nt    = t & 7;               // 8 N-tiles
    const float* Ain = which ? Kin : Q;
    const float* Wm  = which ? Wk  : Wq;
    float*       Cm  = which ? Kt  : Qt;
    v8f c = {};
    c = wmma_f32_tile(Ain + (size_t)mt * 16 * H_, H_, Wm + nt * 16, AD_, H_, c);
    store_tile(Cm + (size_t)mt * 16 * AD_ + nt * 16, AD_, c);
}

// ---------------------------------------------------------------------------
// Kernel 2: scores + softmax. One workgroup per (b,q); thread k owns score k.
// tanh(x) = 1 - 2*rcp(exp2(2*log2e*x)+1)  -> v_exp2_f32 + v_rcp_f32
// A-reduction vectorized to float4: global_load_b128 (4x fewer VMEM issues),
// qt/v read as LDS float4 broadcasts.
// ---------------------------------------------------------------------------
__device__ __forceinline__ float fast_tanh(float x)
{
    float t = __builtin_amdgcn_exp2f(x * 2.8853900817779268f); // 2/ln(2)
    return 1.0f - 2.0f * __builtin_amdgcn_rcpf(t + 1.0f);
}

__global__ __launch_bounds__(512) void k_scores_softmax(
    const float* __restrict__ Qt, const float* __restrict__ Kt,
    const float* __restrict__ vvec, float* __restrict__ Wout)
{
    __shared__ float sh_qt[AD_];
    __shared__ float sh_v[AD_];
    __shared__ float sh_red[16];

    const int bq  = blockIdx.x;          // 0 .. B*SQ-1
    const int b   = bq >> 9;             // / SQ
    const int tid = threadIdx.x;         // == k  (SK == 512 == blockDim)

    if (tid < AD_) {
        sh_qt[tid] = Qt[(size_t)bq * AD_ + tid];
        sh_v[tid]  = vvec[tid];
    }
    __syncthreads();

    const float4* __restrict__ kt4 =
        (const float4*)(Kt + ((size_t)b * SK_ + tid) * AD_);   // 512B-aligned row
    const float4* q4 = (const float4*)sh_qt;
    const float4* v4 = (const float4*)sh_v;

    float s = 0.0f;
#pragma unroll
    for (int a = 0; a < AD_ / 4; ++a) {
        float4 kv = kt4[a];      // global_load_b128
        float4 qv = q4[a];       // ds_load_b128 (broadcast)
        float4 vw = v4[a];       // ds_load_b128 (broadcast)
        s += fast_tanh(qv.x + kv.x) * vw.x;
        s += fast_tanh(qv.y + kv.y) * vw.y;
        s += fast_tanh(qv.z + kv.z) * vw.z;
        s += fast_tanh(qv.w + kv.w) * vw.w;
    }

    // ---- block max (wave32 shuffle + LDS across 16 waves) ----
    float m = s;
#pragma unroll
    for (int off = 16; off > 0; off >>= 1)
        m = fmaxf(m, __shfl_xor(m, off));
    const int wid = tid >> 5;
    if ((tid & 31) == 0) sh_red[wid] = m;
    __syncthreads();
    if (tid < 32) {
        float mm = (tid < 16) ? sh_red[tid] : -__builtin_inff();
#pragma unroll
        for (int off = 8; off > 0; off >>= 1)
            mm = fmaxf(mm, __shfl_xor(mm, off));
        if (tid == 0) sh_red[0] = mm;
    }
    __syncthreads();
    m = sh_red[0];

    float e = __builtin_amdgcn_exp2f((s - m) * 1.4426950408889634f);

    // ---- block sum ----
    float sum = e;
#pragma unroll
    for (int off = 16; off > 0; off >>= 1)
        sum += __shfl_xor(sum, off);
    __syncthreads();                       // everyone done reading sh_red[0]
    if ((tid & 31) == 0) sh_red[wid] = sum;
    __syncthreads();
    if (tid < 32) {
        float ss = (tid < 16) ? sh_red[tid] : 0.0f;
#pragma unroll
        for (int off = 8; off > 0; off >>= 1)
            ss += __shfl_xor(ss, off);
        if (tid == 0) sh_red[0] = ss;
    }
    __syncthreads();

    Wout[(size_t)bq * SK_ + tid] = e / sh_red[0];
}

// ---------------------------------------------------------------------------
// Kernel 3: out_v[b] = weights[b] @ V[b]   (per batch: M=512, N=256, K=512)
// ---------------------------------------------------------------------------
__global__ __launch_bounds__(128) void k_av(
    const float* __restrict__ Wt, const float* __restrict__ V,
    float* __restrict__ Ov)
{
    const int wave = (blockIdx.x * blockDim.x + threadIdx.x) >> 5;
    const int b    = wave >> 9;            // 512 tiles per batch
    const int t    = wave & 511;
    const int mt   = t >> 4;               // 32 M-tiles
    const int nt   = t & 15;               // 16 N-tiles
    const float* Abase = Wt + (size_t)b * SQ_ * SK_ + (size_t)mt * 16 * SK_;
    const float* Bbase = V  + (size_t)b * SK_ * H_  + nt * 16;
    v8f c = {};
    c = wmma_f32_tile(Abase, SK_, Bbase, H_, SK_, c);
    store_tile(Ov + (size_t)b * SQ_ * H_ + (size_t)mt * 16 * H_ + nt * 16, H_, c);
}

// ---------------------------------------------------------------------------
// Kernel 4: out = out_v @ W_o + b_o   (M=2048, N=256, K=256), bias in accum
// ---------------------------------------------------------------------------
__global__ __launch_bounds__(128) void k_out(
    const float* __restrict__ Ov, const float* __restrict__ Wo,
    const float* __restrict__ bo, float* __restrict__ Out)
{
    const int wave = (blockIdx.x * blockDim.x + threadIdx.x) >> 5;
    const int mt   = wave >> 4;            // 128 M-tiles
    const int nt   = wave & 15;            // 16 N-tiles
    const int lane = threadIdx.x & 31;
    const float bias = bo[nt * 16 + (lane & 15)];   // bias depends on column only
    v8f c;
#pragma unroll
    for (int i = 0; i < 8; ++i) c[i] = bias;
    c = wmma_f32_tile(Ov + (size_t)mt * 16 * H_, H_, Wo + nt * 16, H_, H_, c);
    store_tile(Out + (size_t)mt * 16 * H_ + nt * 16, H_, c);
}

// ---------------------------------------------------------------------------
extern "C" void kernel_launch(void* const* d_in, const int* in_sizes, int n_in,
                              void* d_out, int out_size, void* d_ws, size_t ws_size,
                              hipStream_t stream)
{
    const float* Q  = (const float*)d_in[0];
    const float* K  = (const float*)d_in[1];
    const float* V  = (const float*)d_in[2];
    const float* Wq = (const float*)d_in[3];
    const float* Wk = (const float*)d_in[4];
    const float* vv = (const float*)d_in[5];
    const float* Wo = (const float*)d_in[6];
    const float* bo = (const float*)d_in[7];

    float* out     = (float*)d_out;                      // (B,SQ,H)
    float* weights = out + (size_t)B_ * SQ_ * H_;        // (B,SQ,SK) — 2nd output

    float* Qt = (float*)d_ws;                            // (B*SQ, A)  1 MB
    float* Kt = Qt + (size_t)B_ * SQ_ * AD_;             // (B*SK, A)  1 MB
    float* Ov = Kt + (size_t)B_ * SK_ * AD_;             // (B*SQ, H)  2 MB

    // 1) projections: 2048 tile-waves, 4 waves/block
    k_project<<<512, 128, 0, stream>>>(Q, K, Wq, Wk, Qt, Kt);
    // 2) scores + softmax: one block per (b,q), writes weights output directly
    k_scores_softmax<<<B_ * SQ_, 512, 0, stream>>>(Qt, Kt, vv, weights);
    // 3) attention * V: 2048 tile-waves
    k_av<<<512, 128, 0, stream>>>(weights, V, Ov);
    // 4) output projection + bias: 2048 tile-waves
    k_out<<<512, 128, 0, stream>>>(Ov, Wo, bo, out);
}